// GCN_30751965839417
// MI455X (gfx1250) — compile-verified
//
#include <hip/hip_runtime.h>
#include <hip/hip_bf16.h>

typedef __attribute__((ext_vector_type(2))) float v2f;
typedef __attribute__((ext_vector_type(8))) float v8f;

#define N_NODES 100000
#define N_EDGES 1600000
#define N_FEAT  256
#define HIDDEN  128
#define N_CLS   40

// ---------------- degree / normalization ----------------

__global__ void k_init_deg(float* __restrict__ deg, int n) {
    int i = blockIdx.x * blockDim.x + threadIdx.x;
    if (i < n) deg[i] = 1.0f;   // self loop
}

__global__ void k_deg_scatter(const int* __restrict__ dst, float* __restrict__ deg, int e) {
    int i = blockIdx.x * blockDim.x + threadIdx.x;
    if (i < e) atomicAdd(&deg[dst[i]], 1.0f);
}

__global__ void k_rsqrt(float* __restrict__ deg, int n) {
    int i = blockIdx.x * blockDim.x + threadIdx.x;
    if (i < n) deg[i] = rsqrtf(deg[i]);   // deg buffer now holds dinv
}

// ---------------- GEMM1: h1 = X @ W1  (100000x256 @ 256x128), fp32 WMMA ----------------
// grid = (6250, 2), block = 128 (4 waves); wave w handles N-tile (blockIdx.y*4 + w)

__global__ void k_gemm1(const float* __restrict__ X, const float* __restrict__ W,
                        float* __restrict__ H) {
    const int lane = threadIdx.x & 31;
    const int wave = threadIdx.x >> 5;
    const int row0 = blockIdx.x * 16;
    const int col0 = (blockIdx.y * 4 + wave) * 16;

    const int m  = lane & 15;             // A row / D column index source
    const int kk = (lane >> 4) * 2;       // K sub-pair per lane half
    const float* arow = X + (size_t)(row0 + m) * N_FEAT;
    const float* bcol = W + col0 + (lane & 15);

    v8f c = {};
    #pragma unroll 4
    for (int k = 0; k < N_FEAT; k += 4) {
        const int ka = k + kk;
        v2f a = *(const v2f*)(arow + ka);             // A[m][ka], A[m][ka+1]
        v2f b;
        b.x = bcol[(size_t)(ka    ) * HIDDEN];        // B[ka  ][n]
        b.y = bcol[(size_t)(ka + 1) * HIDDEN];        // B[ka+1][n]
        c = __builtin_amdgcn_wmma_f32_16x16x4_f32(false, a, false, b,
                                                  (short)0, c, false, false);
    }

    // D layout: VGPR r -> M = r + 8*(lane>=16), N = lane&15
    const int mbase = row0 + (lane >> 4) * 8;
    float* o = H + (size_t)mbase * HIDDEN + col0 + (lane & 15);
    #pragma unroll
    for (int r = 0; r < 8; ++r) o[(size_t)r * HIDDEN] = c[r];
}

// ---------------- layer-1 self loop + scatter ----------------

__global__ void k_selfloop1(const float* __restrict__ h1, const float* __restrict__ dinv,
                            float* __restrict__ acc, int n) {
    int t = blockIdx.x * blockDim.x + threadIdx.x;
    if (t >= n * 32) return;
    int i = t >> 5, j = (t & 31) * 4;
    float sl = dinv[i] * dinv[i];
    float4 hv = *(const float4*)(h1 + (size_t)i * HIDDEN + j);
    float4 ov = make_float4(hv.x * sl, hv.y * sl, hv.z * sl, hv.w * sl);
    *(float4*)(acc + (size_t)i * HIDDEN + j) = ov;
}

__global__ void k_scatter1(const int* __restrict__ src, const int* __restrict__ dst,
                           const float* __restrict__ dinv, const float* __restrict__ h1,
                           float* __restrict__ acc, int e) {
    long long t = (long long)blockIdx.x * blockDim.x + threadIdx.x;
    if (t >= (long long)e * 32) return;
    int ed = (int)(t >> 5), j = ((int)t & 31) * 4;
    int s = src[ed], d = dst[ed];
    float norm = dinv[s] * dinv[d];
    float4 hv = *(const float4*)(h1 + (size_t)s * HIDDEN + j);
    float* o = acc + (size_t)d * HIDDEN + j;
    atomicAdd(o + 0, hv.x * norm);
    atomicAdd(o + 1, hv.y * norm);
    atomicAdd(o + 2, hv.z * norm);
    atomicAdd(o + 3, hv.w * norm);
}

// ---------------- GEMM2: h2 = relu(acc1 + b1) @ W2  (100000x128 @ 128x40) ----------------
// grid = 6250, block = 96 (3 waves); wave w handles N-tile w (cols 0..47, masked at 40)

__global__ void k_gemm2(const float* __restrict__ A, const float* __restrict__ b1,
                        const float* __restrict__ W2, float* __restrict__ H2) {
    const int lane = threadIdx.x & 31;
    const int wave = threadIdx.x >> 5;
    const int row0 = blockIdx.x * 16;
    const int col0 = wave * 16;
    const int n    = col0 + (lane & 15);
    const int m    = lane & 15;
    const int kk   = (lane >> 4) * 2;
    const float* arow = A + (size_t)(row0 + m) * HIDDEN;
    const bool nvalid = (n < N_CLS);

    v8f c = {};
    #pragma unroll 4
    for (int k = 0; k < HIDDEN; k += 4) {
        const int ka = k + kk;
        v2f a  = *(const v2f*)(arow + ka);
        v2f bb = *(const v2f*)(b1 + ka);
        a.x = fmaxf(a.x + bb.x, 0.0f);                // fused bias + ReLU
        a.y = fmaxf(a.y + bb.y, 0.0f);
        v2f b;
        b.x = nvalid ? W2[(size_t)(ka    ) * N_CLS + n] : 0.0f;
        b.y = nvalid ? W2[(size_t)(ka + 1) * N_CLS + n] : 0.0f;
        c = __builtin_amdgcn_wmma_f32_16x16x4_f32(false, a, false, b,
                                                  (short)0, c, false, false);
    }

    if (nvalid) {
        const int mbase = row0 + (lane >> 4) * 8;
        float* o = H2 + (size_t)mbase * N_CLS + n;
        #pragma unroll
        for (int r = 0; r < 8; ++r) o[(size_t)r * N_CLS] = c[r];
    }
}

// ---------------- layer-2 self loop (+bias, init d_out) + scatter ----------------

__global__ void k_selfinit2(const float* __restrict__ h2, const float* __restrict__ dinv,
                            const float* __restrict__ b2, float* __restrict__ out, int n) {
    int t = blockIdx.x * blockDim.x + threadIdx.x;
    if (t >= n * 10) return;
    int i = t / 10, j = (t - i * 10) * 4;
    float sl = dinv[i] * dinv[i];
    float4 hv = *(const float4*)(h2 + (size_t)i * N_CLS + j);
    float4 bv = *(const float4*)(b2 + j);
    float4 ov = make_float4(hv.x * sl + bv.x, hv.y * sl + bv.y,
                            hv.z * sl + bv.z, hv.w * sl + bv.w);
    *(float4*)(out + (size_t)i * N_CLS + j) = ov;
}

__global__ void k_scatter2(const int* __restrict__ src, const int* __restrict__ dst,
                           const float* __restrict__ dinv, const float* __restrict__ h2,
                           float* __restrict__ out, int e) {
    long long t = (long long)blockIdx.x * blockDim.x + threadIdx.x;
    if (t >= (long long)e * 10) return;
    int ed = (int)(t / 10), j = ((int)t - ed * 10) * 4;
    int s = src[ed], d = dst[ed];
    float norm = dinv[s] * dinv[d];
    float4 hv = *(const float4*)(h2 + (size_t)s * N_CLS + j);
    float* o = out + (size_t)d * N_CLS + j;
    atomicAdd(o + 0, hv.x * norm);
    atomicAdd(o + 1, hv.y * norm);
    atomicAdd(o + 2, hv.z * norm);
    atomicAdd(o + 3, hv.w * norm);
}

// ---------------- launch ----------------

extern "C" void kernel_launch(void* const* d_in, const int* in_sizes, int n_in,
                              void* d_out, int out_size, void* d_ws, size_t ws_size,
                              hipStream_t stream) {
    const float* x  = (const float*)d_in[0];
    const int*   ei = (const int*)  d_in[1];
    const float* W1 = (const float*)d_in[2];
    const float* b1 = (const float*)d_in[3];
    const float* W2 = (const float*)d_in[4];
    const float* b2 = (const float*)d_in[5];
    float* out = (float*)d_out;

    // workspace layout (floats)
    float* ws   = (float*)d_ws;
    float* deg  = ws;                       // 100000  (deg -> dinv in place)
    float* h1   = ws + 102400;              // 12.8M
    float* acc1 = h1 + (size_t)N_NODES * HIDDEN;   // 12.8M
    float* h2   = acc1 + (size_t)N_NODES * HIDDEN; // 4.0M

    const int* src = ei;
    const int* dst = ei + N_EDGES;

    // normalization
    k_init_deg   <<<(N_NODES + 255) / 256, 256, 0, stream>>>(deg, N_NODES);
    k_deg_scatter<<<(N_EDGES + 255) / 256, 256, 0, stream>>>(dst, deg, N_EDGES);
    k_rsqrt      <<<(N_NODES + 255) / 256, 256, 0, stream>>>(deg, N_NODES);

    // layer 1
    k_gemm1<<<dim3(N_NODES / 16, 2), 128, 0, stream>>>(x, W1, h1);
    k_selfloop1<<<(N_NODES * 32 + 255) / 256, 256, 0, stream>>>(h1, deg, acc1, N_NODES);
    {
        long long threads = (long long)N_EDGES * 32;
        k_scatter1<<<(unsigned)((threads + 255) / 256), 256, 0, stream>>>(src, dst, deg, h1, acc1, N_EDGES);
    }

    // layer 2 (bias+ReLU fused into GEMM2 A-load)
    k_gemm2<<<N_NODES / 16, 96, 0, stream>>>(acc1, b1, W2, h2);
    k_selfinit2<<<(N_NODES * 10 + 255) / 256, 256, 0, stream>>>(h2, deg, b2, out, N_NODES);
    {
        long long threads = (long long)N_EDGES * 10;
        k_scatter2<<<(unsigned)((threads + 255) / 256), 256, 0, stream>>>(src, dst, deg, h2, out, N_EDGES);
    }
}